// QPSKModulator_14207751815948
// MI455X (gfx1250) — compile-verified
//
#include <hip/hip_runtime.h>

// QPSK modulator + RRC polyphase pulse shaping for MI455X (gfx1250, wave32).
//
// Polyphase identity (sps=16, K=128, pad_l=63, offset=8):
//   y[16s+p] = sum_{t=0..7} rrc[j0(p)+16t] * sym[s+t-d(p)],
//   j0(p) = (71-p) mod 16  ( = 7-p for p<8, 23-p for p>=8 ),
//   d(p)  = 4 for p<8, 3 for p>=8.
// With u = t + (4-d(p)) both halves become one GEMM tile:
//   Y[16x16] = A[16x12] x B[12x16]   (K padded 9 -> 12)
//   A[s_local][u] = sym[s_base + s_local + u - 4]
//   B[u][p]       = sqrt(16) * rrc[...] per the phase table, 0 elsewhere.
// Computed with 3x V_WMMA_F32_16X16X4_F32 per channel per tile.
//
// LDS is SoA (separate re/im planes) so each A fragment {sym[u], sym[u+1]}
// is one ds_load_2addr_b32 straight into an even VGPR pair (no repack moves).

typedef __attribute__((ext_vector_type(2))) float v2f;
typedef __attribute__((ext_vector_type(8))) float v8f;

#define NS        8192      // symbols per batch row
#define NB        16384     // bits per batch row
#define UP_LEN    131072    // output samples per row (per channel)
#define NROWS     256
#define TILES_PER_WAVE 8
#define SYMS_PER_WAVE  (TILES_PER_WAVE * 16)   // 128
#define LDS_SYMS       (SYMS_PER_WAVE + 24)    // window [s0-4, s0+147]
#define LDS_PAD        160
#define WAVES_PER_BLOCK 8

__global__ __launch_bounds__(256)
void qpsk_rrc_wmma(const int* __restrict__ bits,
                   const float* __restrict__ rrc,
                   float* __restrict__ out) {
  __shared__ float smre[WAVES_PER_BLOCK][LDS_PAD];
  __shared__ float smim[WAVES_PER_BLOCK][LDS_PAD];

  const int lane      = threadIdx.x & 31;
  const int waveInBlk = threadIdx.x >> 5;
  const int wave      = blockIdx.x * WAVES_PER_BLOCK + waveInBlk;
  const int runsPerRow = NS / SYMS_PER_WAVE;          // 64
  const int row = wave / runsPerRow;
  const int s0  = (wave % runsPerRow) * SYMS_PER_WAVE;

  const int p    = lane & 15;   // phase (B/D column N)
  const int half = lane >> 4;   // lane-half selects K rows / D M-half

  // --- B (tap) matrix fragments, K=12 zero-padded, scaled by sqrt(sps)=4.
  // Assumed f32 B-fragment layout mirrors A (ISA 7.12.2): for each x4 chunk
  // k0=4i, VGPR j holds B[k0 + 2*half + j][N=lane&15].
  float bt[3][2];
#pragma unroll
  for (int i = 0; i < 3; ++i) {
#pragma unroll
    for (int j = 0; j < 2; ++j) {
      const int u = 4 * i + 2 * half + j;
      float v = 0.0f;
      if (p < 8) {
        if (u < 8) v = 4.0f * rrc[(7 - p) + 16 * u];
      } else {
        if (u >= 1 && u <= 8) v = 4.0f * rrc[(23 - p) + 16 * (u - 1)];
      }
      bt[i][j] = v;
    }
  }
  const v2f bf0 = {bt[0][0], bt[0][1]};
  const v2f bf1 = {bt[1][0], bt[1][1]};
  const v2f bf2 = {bt[2][0], bt[2][1]};

  // --- Decode this wave's symbol window into its private LDS slice (SoA).
  // Gray map: re = (1-2*bits[2k+1])/sqrt2, im = (1-2*bits[2k])/sqrt2.
  const int* __restrict__ brow = bits + (long)row * NB;
  const float inv_sqrt2 = 0.70710678118654752440f;
  for (int i = lane; i < LDS_SYMS; i += 32) {
    const int k = s0 - 4 + i;
    float re = 0.0f, im = 0.0f;
    if (k >= 0 && k < NS) {
      const int b0 = brow[2 * k];
      const int b1 = brow[2 * k + 1];
      re = (float)(1 - 2 * b1) * inv_sqrt2;
      im = (float)(1 - 2 * b0) * inv_sqrt2;
    }
    smre[waveInBlk][i] = re;
    smim[waveInBlk][i] = im;
  }
  __syncthreads();

  v2f* __restrict__ o2 = (v2f*)out;  // interleaved (re, im) pairs

  for (int t = 0; t < TILES_PER_WAVE; ++t) {
    // A fragment (16x4 f32, ISA 7.12.2): lane holds M=p, VGPR j = K=2*half+j.
    // LDS index for u is t*16 + M + u; u = 4*i + 2*half + j.
    const int bidx = t * 16 + p + 2 * half;
    const float* __restrict__ lre = &smre[waveInBlk][bidx];
    const float* __restrict__ lim = &smim[waveInBlk][bidx];

    v2f aRe0, aIm0, aRe1, aIm1, aRe2, aIm2;
    aRe0.x = lre[0]; aRe0.y = lre[1];
    aIm0.x = lim[0]; aIm0.y = lim[1];
    aRe1.x = lre[4]; aRe1.y = lre[5];
    aIm1.x = lim[4]; aIm1.y = lim[5];
    aRe2.x = lre[8]; aRe2.y = lre[9];
    aIm2.x = lim[8]; aIm2.y = lim[9];

    v8f cre = {};
    v8f cim = {};
    // Interleave the two accumulator chains for ILP.
    cre = __builtin_amdgcn_wmma_f32_16x16x4_f32(false, aRe0, false, bf0,
                                                (short)0, cre, false, false);
    cim = __builtin_amdgcn_wmma_f32_16x16x4_f32(false, aIm0, false, bf0,
                                                (short)0, cim, false, false);
    cre = __builtin_amdgcn_wmma_f32_16x16x4_f32(false, aRe1, false, bf1,
                                                (short)0, cre, false, false);
    cim = __builtin_amdgcn_wmma_f32_16x16x4_f32(false, aIm1, false, bf1,
                                                (short)0, cim, false, false);
    cre = __builtin_amdgcn_wmma_f32_16x16x4_f32(false, aRe2, false, bf2,
                                                (short)0, cre, false, false);
    cim = __builtin_amdgcn_wmma_f32_16x16x4_f32(false, aIm2, false, bf2,
                                                (short)0, cim, false, false);

    // D layout: VGPR g -> M = g + 8*half, N = p. Output sample
    // n = (s0 + t*16 + M)*16 + p; (re,im) pair is contiguous -> 8B store.
    // NT hint: 256MB stream, written once, keep it out of L2.
    const long nbase = (long)row * UP_LEN + (long)(s0 + t * 16) * 16 + p;
#pragma unroll
    for (int g = 0; g < 8; ++g) {
      v2f v;
      v.x = cre[g];
      v.y = cim[g];
      __builtin_nontemporal_store(v, &o2[nbase + (long)(g + 8 * half) * 16]);
    }
  }
}

extern "C" void kernel_launch(void* const* d_in, const int* in_sizes, int n_in,
                              void* d_out, int out_size, void* d_ws, size_t ws_size,
                              hipStream_t stream) {
  (void)in_sizes; (void)n_in; (void)out_size; (void)d_ws; (void)ws_size;
  const int*   bits = (const int*)d_in[0];
  const float* rrc  = (const float*)d_in[1];
  float*       out  = (float*)d_out;

  const int totalWaves = NROWS * (NS / SYMS_PER_WAVE);   // 16384
  const int blocks     = totalWaves / WAVES_PER_BLOCK;   // 2048
  hipLaunchKernelGGL(qpsk_rrc_wmma, dim3(blocks), dim3(256), 0, stream,
                     bits, rrc, out);
}